// Subtractor4Bit_43860206027272
// MI455X (gfx1250) — compile-verified
//
#include <hip/hip_runtime.h>
#include <cstdint>

// ---------------------------------------------------------------------------
// 4-bit ripple-borrow subtractor over {0.0,1.0}-encoded bits.
// Pure HBM-streaming op: 256 MiB read + 160 MiB write -> ~19 us @ 23.3 TB/s.
// No contraction dimension exists, so WMMA is inapplicable; the gfx1250
// feature that matters is the async global->LDS copy engine (ASYNCcnt),
// which we run as a 4-deep ring to keep ~3 KB/wave in flight (VGPR-free)
// while compute consumes the oldest stage. NT b128 stores keep the
// write-once output stream out of the 192 MB L2 hot set.
// ---------------------------------------------------------------------------

#define TPB 256
#define DEPTH 4           // async ring depth (stages in flight)
#define STAGE_BYTES 4096  // TPB * 16 bytes per buffer

typedef float v4f __attribute__((ext_vector_type(4)));

// addrspace(3) offsets start at 0 within the workgroup's LDS allocation and
// the generic aperture base has zero low 32 bits, so truncating a generic
// pointer to LDS yields the byte address the async engine expects
// (a byte offset added to the wave's LDS base).
__device__ __forceinline__ uint32_t lds_addr32(const void* p) {
  return (uint32_t)(uintptr_t)p;
}

// GLOBAL_LOAD_ASYNC_TO_LDS_B128: per-lane 16B global->LDS, tracked by ASYNCcnt.
__device__ __forceinline__ void async_copy_b128(uint32_t lds_byte_off,
                                                const void* gsrc) {
  asm volatile("global_load_async_to_lds_b128 %0, %1, off"
               :
               : "v"(lds_byte_off), "v"((unsigned long long)gsrc)
               : "memory");
}

// Exact replacement of the float gate network: pack nibbles (elem 3 = LSB),
// subtract, unpack. Inputs are exactly 0.0f or 1.0f.
__device__ __forceinline__ void sub4(v4f a, v4f b, v4f* d, float* brw) {
  unsigned ua = (unsigned)a[3] | ((unsigned)a[2] << 1) |
                ((unsigned)a[1] << 2) | ((unsigned)a[0] << 3);
  unsigned ub = (unsigned)b[3] | ((unsigned)b[2] << 1) |
                ((unsigned)b[1] << 2) | ((unsigned)b[0] << 3);
  unsigned dd = (ua - ub) & 15u;
  *brw = (ua < ub) ? 1.0f : 0.0f;       // borrow-out of the 4-bit subtract
  (*d)[0] = (float)((dd >> 3) & 1u);    // diffs[0] has weight 2^3
  (*d)[1] = (float)((dd >> 2) & 1u);
  (*d)[2] = (float)((dd >> 1) & 1u);
  (*d)[3] = (float)(dd & 1u);           // diffs[3] is the LSB
}

__global__ void __launch_bounds__(TPB)
sub4_kernel(const v4f* __restrict__ A, const v4f* __restrict__ B,
            float* __restrict__ out, int n) {
  // 4-deep ring: 2 arrays * 4 * 4 KB = 32 KB of the WGP's 320 KB LDS.
  __shared__ v4f sA[DEPTH][TPB];
  __shared__ v4f sB[DEPTH][TPB];

  v4f* __restrict__ res = reinterpret_cast<v4f*>(out);     // [n] rows of 4
  float* __restrict__ bor = out + (size_t)4 * (size_t)n;   // [n] borrows

  const int nt = n / TPB;              // full 256-row tiles
  const unsigned tid = threadIdx.x;
  const int g = gridDim.x;
  const int t0 = blockIdx.x;

  // Stages this block processes: tiles t0, t0+g, t0+2g, ...
  const int total = (t0 < nt) ? ((nt - t0 + g - 1) / g) : 0;

  const uint32_t offA0 = lds_addr32(&sA[0][tid]);
  const uint32_t offB0 = lds_addr32(&sB[0][tid]);

  const size_t stepRows = (size_t)g * TPB;     // rows between my stages
  const v4f* ia = A + (size_t)t0 * TPB + tid;  // next stage to ISSUE
  const v4f* ib = B + (size_t)t0 * TPB + tid;

  // Prime: fill up to DEPTH ring slots (2 async ops per wave per stage).
  const int nprime = total < DEPTH ? total : DEPTH;
  for (int d = 0; d < nprime; ++d) {
    const uint32_t ds = (uint32_t)d * STAGE_BYTES;
    async_copy_b128(offA0 + ds, ia);
    async_copy_b128(offB0 + ds, ib);
    ia += stepRows;
    ib += stepRows;
  }

  v4f* pr = res + (size_t)t0 * TPB + tid;      // next stage to STORE
  float* pw = bor + (size_t)t0 * TPB + tid;

  for (int s = 0; s < total; ++s) {
    // Stages still in flight beyond the one we consume: min(DEPTH-1, rem).
    const int rem = total - 1 - s;
    if (rem >= 3)      __builtin_amdgcn_s_wait_asynccnt(6);
    else if (rem == 2) __builtin_amdgcn_s_wait_asynccnt(4);
    else if (rem == 1) __builtin_amdgcn_s_wait_asynccnt(2);
    else               __builtin_amdgcn_s_wait_asynccnt(0);

    const int d = s & (DEPTH - 1);
    v4f a = sA[d][tid];                        // ds_load_b128
    v4f b = sB[d][tid];
    v4f dd;
    float w;
    sub4(a, b, &dd, &w);
    __builtin_nontemporal_store(dd, pr);       // write-once stream: NT
    __builtin_nontemporal_store(w, pw);
    pr += stepRows;
    pw += stepRows;

    // Refill the slot we just drained (ds_load waited -> no WAR hazard).
    if (s + DEPTH < total) {
      const uint32_t ds = (uint32_t)d * STAGE_BYTES;
      async_copy_b128(offA0 + ds, ia);
      async_copy_b128(offB0 + ds, ib);
      ia += stepRows;
      ib += stepRows;
    }
  }

  // Tail rows (none for N = 2^23, but keep it general), direct loads.
  for (int r = nt * TPB + blockIdx.x * TPB + tid; r < n;
       r += g * TPB) {
    v4f a = A[r];
    v4f b = B[r];
    v4f dd;
    float w;
    sub4(a, b, &dd, &w);
    __builtin_nontemporal_store(dd, &res[r]);
    __builtin_nontemporal_store(w, &bor[r]);
  }
}

extern "C" void kernel_launch(void* const* d_in, const int* in_sizes, int n_in,
                              void* d_out, int out_size, void* d_ws,
                              size_t ws_size, hipStream_t stream) {
  (void)n_in; (void)out_size; (void)d_ws; (void)ws_size;
  const v4f* A = (const v4f*)d_in[0];
  const v4f* B = (const v4f*)d_in[1];
  float* out = (float*)d_out;
  const int n = in_sizes[0] / 4;       // rows

  // 32768 tiles for N=2^23; 4096 blocks -> 8 pipeline stages per block,
  // plenty to reach steady state with a 4-deep ring.
  int tiles = n / TPB;
  int blocks = tiles > 0 ? (tiles < 4096 ? tiles : 4096) : 1;
  sub4_kernel<<<blocks, TPB, 0, stream>>>(A, B, out, n);
}